// HashGridEmbedder_76467597738034
// MI455X (gfx1250) — compile-verified
//
#include <hip/hip_runtime.h>
#include <stdint.h>

// ---------------------------------------------------------------------------
// HashGrid encoder for MI455X (gfx1250, wave32).
//
// Per point: 16 levels x 8 corners, hash -> gather float2 -> trilinear blend.
// Embedding table (42 MB) is L2-resident (192 MB L2) -> gathers use default
// temporal hint. Output (140 MB, write-once) uses non-temporal b128 stores so
// it does not evict the table from L2. Output staged through LDS for fully
// coalesced write-back.
//
// VALU diet:
//  - 3 IEEE divisions total (not 48): g_l = ((p+0.75)/1.5) * res_l
//  - hashed levels: hash pre-shifted by 3 -> per corner just XOR+AND, result
//    is the byte voffset; level offset folded into SGPR base pointer
//  - dense levels: h%T via 32-bit decomposition (hi = h>>32 < 64, shared
//    across the 4 (y,z) combos)
//  - feature-pair accumulation via v_pk_fma_f32 (op_sel scalar broadcast)
// ---------------------------------------------------------------------------

typedef float v2f __attribute__((ext_vector_type(2)));
typedef float v4f __attribute__((ext_vector_type(4)));

namespace {

constexpr int NLVL = 16;
constexpr int NOUT = 35;              // 3 (xyz) + 16*2
constexpr int BLK  = 256;             // 8 wave32 per workgroup

// res_i = floor(16 * 2^(i/3))  (per-level scale = 32^(1/15) = 2^(1/3))
constexpr int kRes[NLVL] = {16, 20, 25, 32, 40, 50, 64, 80,
                            101, 128, 161, 203, 256, 322, 406, 512};

// tsize_i = min(2^19, (res_i + 1)^3)
constexpr uint32_t kTsize[NLVL] = {
    4913u, 9261u, 17576u, 35937u, 68921u, 132651u, 274625u,
    524288u, 524288u, 524288u, 524288u, 524288u, 524288u, 524288u, 524288u, 524288u};

// cumulative offsets into the embedding table
constexpr uint32_t kOff[NLVL] = {
    0u, 4913u, 14174u, 31750u, 67687u, 136608u, 269259u, 543884u,
    1068172u, 1592460u, 2116748u, 2641036u, 3165324u, 3689612u,
    4213900u, 4738188u};

constexpr uint64_t P1 = 2654435761ull;   // prime for y
constexpr uint64_t P2 = 805459861ull;    // prime for z
// pre-shifted (x8) primes for the power-of-two (hashed) levels: only the low
// 22 bits survive the mask, so wrapping u32 arithmetic is exact.
constexpr uint32_t P1S = (uint32_t)(P1 << 3);
constexpr uint32_t P2S = (uint32_t)(P2 << 3);

} // namespace

static __device__ __forceinline__ v2f splat(float v) { return (v2f){v, v}; }

__global__ __launch_bounds__(BLK) void hashgrid_encode_kernel(
    const float* __restrict__ xyz,       // (B,3)
    const float* __restrict__ emb,       // (N_total, 2)
    float* __restrict__ out,             // (B,35)
    int B)
{
    __shared__ __align__(16) float tile[BLK * NOUT];   // 35840 B

    const int tid  = (int)threadIdx.x;
    const int base = (int)blockIdx.x * BLK;
    const int pi = base + tid;
    const int pc = (pi < B) ? pi : (B - 1);            // clamp tail (writeback guarded)

    const float x = xyz[3 * pc + 0];
    const float y = xyz[3 * pc + 1];
    const float z = xyz[3 * pc + 2];

    float* __restrict__ row = &tile[tid * NOUT];
    row[0] = x; row[1] = y; row[2] = z;                // include_input

    // g_l = ((p - min_c) / 1.5) * res_l : one IEEE div per dim, then a
    // per-level multiply. Bit-exact vs reference for power-of-2 res levels;
    // <=1-2 ulp otherwise (interpolation is continuous in g -> negligible).
    const float sx = (x + 0.75f) / 1.5f;
    const float sy = (y + 0.75f) / 1.5f;
    const float sz = (z + 0.75f) / 1.5f;

    // Level-independent validity: s in [0,1) <=> 0 <= floor(s*res) < res for
    // every level (exact away from the boundary; inputs carry a 0.1% margin).
    const bool vall = (sx >= 0.0f) & (sx < 1.0f) &
                      (sy >= 0.0f) & (sy < 1.0f) &
                      (sz >= 0.0f) & (sz < 1.0f);

#pragma unroll
    for (int l = 0; l < NLVL; ++l) {
        const int      res = kRes[l];
        const uint32_t T   = kTsize[l];
        // level offset folded into a uniform (SGPR) base pointer
        const char* __restrict__ baseL = (const char*)emb + (uint64_t)kOff[l] * 8u;

        const float gx = sx * (float)res;
        const float gy = sy * (float)res;
        const float gz = sz * (float)res;

        const int ix = (int)floorf(gx);
        const int iy = (int)floorf(gy);
        const int iz = (int)floorf(gz);

        const int cx = ix < 0 ? 0 : (ix > res - 1 ? res - 1 : ix);   // v_med3
        const int cy = iy < 0 ? 0 : (iy > res - 1 ? res - 1 : iy);
        const int cz = iz < 0 ? 0 : (iz > res - 1 ? res - 1 : iz);

        const float tx = gx - (float)cx;
        const float ty = gy - (float)cy;
        const float tz = gz - (float)cz;
        const float wx0 = 1.0f - tx, wx1 = tx;
        const float wy0 = 1.0f - ty, wy1 = ty;
        const float wz0 = 1.0f - tz, wz1 = tz;

        uint32_t off[8];   // byte offsets from baseL
        if ((T & (T - 1u)) == 0u) {
            // hashed levels: tsize == 2^19. Everything pre-shifted by 3 so the
            // masked XOR is directly the byte offset: per corner = XOR + AND.
            const uint32_t maskS = (T - 1u) << 3;                  // 0x3FFFF8
            const uint32_t hy0 = (uint32_t)cy * P1S;
            const uint32_t hy1 = hy0 + P1S;
            const uint32_t hz0 = (uint32_t)cz * P2S;
            const uint32_t hz1 = hz0 + P2S;
            const uint32_t m00 = hy0 ^ hz0;    // (oy,oz)=(0,0)
            const uint32_t m01 = hy0 ^ hz1;
            const uint32_t m10 = hy1 ^ hz0;
            const uint32_t m11 = hy1 ^ hz1;
            const uint32_t x0 = (uint32_t)cx << 3, x1 = x0 + 8u;
            off[0] = (m00 ^ x0) & maskS;
            off[1] = (m01 ^ x0) & maskS;
            off[2] = (m10 ^ x0) & maskS;
            off[3] = (m11 ^ x0) & maskS;
            off[4] = (m00 ^ x1) & maskS;
            off[5] = (m01 ^ x1) & maskS;
            off[6] = (m10 ^ x1) & maskS;
            off[7] = (m11 ^ x1) & maskS;
        } else {
            // dense levels (res <= 64): h < 2^41 and hi = h>>32 < 64.
            // h % T = ((hi*(2^32 % T)) % T + lo % T) mod T, hi part shared
            // across the 4 (y,z) combos (x only touches lo).
            const uint32_t c = (uint32_t)(0x100000000ull % (uint64_t)T); // constexpr
            const uint64_t hy0 = (uint64_t)(uint32_t)cy * P1;
            const uint64_t hy1 = hy0 + P1;
            const uint64_t hz0 = (uint64_t)(uint32_t)cz * P2;
            const uint64_t hz1 = hz0 + P2;
            const uint64_t m00 = hy0 ^ hz0;
            const uint64_t m01 = hy0 ^ hz1;
            const uint64_t m10 = hy1 ^ hz0;
            const uint64_t m11 = hy1 ^ hz1;
            const uint32_t hr00 = ((uint32_t)(m00 >> 32) * c) % T;
            const uint32_t hr01 = ((uint32_t)(m01 >> 32) * c) % T;
            const uint32_t hr10 = ((uint32_t)(m10 >> 32) * c) % T;
            const uint32_t hr11 = ((uint32_t)(m11 >> 32) * c) % T;
            const uint32_t lo00 = (uint32_t)m00, lo01 = (uint32_t)m01;
            const uint32_t lo10 = (uint32_t)m10, lo11 = (uint32_t)m11;
            const uint32_t x0 = (uint32_t)cx, x1 = x0 + 1u;

            auto red = [T](uint32_t hr, uint32_t lo) -> uint32_t {
                uint32_t s = hr + lo % T;                  // s < 2T
                return (s >= T ? s - T : s) << 3;          // byte offset
            };
            off[0] = red(hr00, lo00 ^ x0);
            off[1] = red(hr01, lo01 ^ x0);
            off[2] = red(hr10, lo10 ^ x0);
            off[3] = red(hr11, lo11 ^ x0);
            off[4] = red(hr00, lo00 ^ x1);
            off[5] = red(hr01, lo01 ^ x1);
            off[6] = red(hr10, lo10 ^ x1);
            off[7] = red(hr11, lo11 ^ x1);
        }

        // Issue all 8 gathers (global_load_b64, saddr + voffset) before the
        // FMA chain so the compiler can cover L2 latency within the level.
        v2f e[8];
#pragma unroll
        for (int k = 0; k < 8; ++k) e[k] = *(const v2f*)(baseL + off[k]);

        // corner order: k = (ox<<2)|(oy<<1)|oz  (z fastest, matches reference)
        // weight pairs over z -> v_pk_mul_f32; accumulation over the feature
        // pair -> v_pk_fma_f32 with scalar op_sel broadcast.
        const float wxy00 = wx0 * wy0, wxy01 = wx0 * wy1;
        const float wxy10 = wx1 * wy0, wxy11 = wx1 * wy1;
        const v2f wzv  = {wz0, wz1};
        const v2f wp0  = splat(wxy00) * wzv;   // {w0, w1}
        const v2f wp1  = splat(wxy01) * wzv;   // {w2, w3}
        const v2f wp2  = splat(wxy10) * wzv;   // {w4, w5}
        const v2f wp3  = splat(wxy11) * wzv;   // {w6, w7}

        v2f acc = {0.0f, 0.0f};
        acc = __builtin_elementwise_fma(splat(wp0[0]), e[0], acc);
        acc = __builtin_elementwise_fma(splat(wp0[1]), e[1], acc);
        acc = __builtin_elementwise_fma(splat(wp1[0]), e[2], acc);
        acc = __builtin_elementwise_fma(splat(wp1[1]), e[3], acc);
        acc = __builtin_elementwise_fma(splat(wp2[0]), e[4], acc);
        acc = __builtin_elementwise_fma(splat(wp2[1]), e[5], acc);
        acc = __builtin_elementwise_fma(splat(wp3[0]), e[6], acc);
        acc = __builtin_elementwise_fma(splat(wp3[1]), e[7], acc);

        if (!vall) acc = (v2f){0.0f, 0.0f};

        row[3 + 2 * l + 0] = acc[0];
        row[3 + 2 * l + 1] = acc[1];
    }

    __syncthreads();

    // Coalesced non-temporal write-back: 256*35 = 8960 floats = 2240 v4f.
    // NT hint keeps the 42 MB embedding table resident in the 192 MB L2.
    const int outN  = B * NOUT;                 // 36,700,160 -> fits int32
    const int gbase = base * NOUT;              // multiple of 8960 -> 16B aligned
    const v4f* __restrict__ t4 = (const v4f*)tile;

#pragma unroll
    for (int j = 0; j < 9; ++j) {
        const int n = j * BLK + tid;            // v4f index within block chunk
        if (n < (BLK * NOUT) / 4) {
            const int g = gbase + n * 4;        // float index, multiple of 4
            if (g + 3 < outN) {
                __builtin_nontemporal_store(t4[n], (v4f*)(out + g));
            } else {
                const v4f v = t4[n];
                if (g + 0 < outN) out[g + 0] = v[0];
                if (g + 1 < outN) out[g + 1] = v[1];
                if (g + 2 < outN) out[g + 2] = v[2];
                if (g + 3 < outN) out[g + 3] = v[3];
            }
        }
    }
}

extern "C" void kernel_launch(void* const* d_in, const int* in_sizes, int n_in,
                              void* d_out, int out_size, void* d_ws, size_t ws_size,
                              hipStream_t stream) {
    (void)n_in; (void)out_size; (void)d_ws; (void)ws_size;
    const float* xyz = (const float*)d_in[0];   // (B,3) f32
    const float* emb = (const float*)d_in[1];   // (N_total,2) f32
    float* out = (float*)d_out;                 // (B,35) f32

    const int B = in_sizes[0] / 3;
    const int blocks = (B + BLK - 1) / BLK;
    hipLaunchKernelGGL(hashgrid_encode_kernel, dim3(blocks), dim3(BLK), 0, stream,
                       xyz, emb, out, B);
}